// FactoredYiJingQuantizer_75754633167491
// MI455X (gfx1250) — compile-verified
//
#include <hip/hip_runtime.h>
#include <stdint.h>

// out = x + (tanh(x * 2/TEMP) - x)  — exact algebraic collapse of the
// factored 8-trigram soft-quantizer: softmax weights over {±1}^3 factor
// per dimension, so the trigram-weighted sum is tanh(2 z_d / T) per dim.
#define TEMP 0.3f
#define KSCALE (2.0f / TEMP)

typedef __attribute__((ext_vector_type(4))) float v4f;

__device__ __forceinline__ float soft_quant_dim(float z) {
#if __has_builtin(__builtin_amdgcn_tanhf)
    float q = __builtin_amdgcn_tanhf(z * KSCALE);   // gfx1250 v_tanh_f32
#else
    float q = tanhf(z * KSCALE);
#endif
    return z + (q - z);   // reproduce reference's final x + (quantized - x)
}

// CDNA5 async copy: one 16B element per lane, global -> this wave's LDS slot.
// Tracked by ASYNCcnt; non-temporal (805MB stream > 192MB L2).
__device__ __forceinline__ void async_load16(void* lds_dst, const v4f* gsrc) {
    uint32_t lds_off = (uint32_t)(uintptr_t)lds_dst;  // low 32 bits = LDS byte offset
    asm volatile("global_load_async_to_lds_b128 %0, %1, off th:TH_LOAD_NT"
                 :: "v"(lds_off), "v"(gsrc)
                 : "memory");
}

// Streaming kernel: per-wave double-buffered async staging into LDS,
// 2x B128 per lane per stage (1KB/wave/stage, up to 2KB in flight/wave).
// Waves only read LDS they themselves loaded -> s_wait_asynccnt is the only
// synchronization needed (no workgroup barriers). Async ops complete in
// order, so waiting asynccnt<=2 retires exactly the current stage.
__global__ void __launch_bounds__(256)
yijing_tanh_stream(const v4f* __restrict__ x, v4f* __restrict__ out,
                   int iters, long T /* total threads */) {
    __shared__ v4f tile[2][2][256];   // 16 KB: 2 stages x 2 elems x 256 threads
    const int  tid = threadIdx.x;
    long i = (long)blockIdx.x * blockDim.x + tid;   // float4 index, elem 0
    const long step = 2 * T;                        // float4s per stage

    // Prologue: fill stage 0 (indices i and i+T are < 2T <= n4 when iters>=1).
    async_load16(&tile[0][0][tid], x + i);
    async_load16(&tile[0][1][tid], x + i + T);
    int buf = 0;

    for (int it = 0; it < iters; ++it) {            // iters is wave-uniform
        long inext = i + step;
        if (it + 1 < iters) {
            async_load16(&tile[buf ^ 1][0][tid], x + inext);
            async_load16(&tile[buf ^ 1][1][tid], x + inext + T);
            asm volatile("s_wait_asynccnt 0x2" ::: "memory"); // current stage done
        } else {
            asm volatile("s_wait_asynccnt 0x0" ::: "memory");
        }

        v4f a = tile[buf][0][tid];                  // ds_load_b128
        v4f b = tile[buf][1][tid];
        v4f ra, rb;
        ra.x = soft_quant_dim(a.x);  ra.y = soft_quant_dim(a.y);
        ra.z = soft_quant_dim(a.z);  ra.w = soft_quant_dim(a.w);
        rb.x = soft_quant_dim(b.x);  rb.y = soft_quant_dim(b.y);
        rb.z = soft_quant_dim(b.z);  rb.w = soft_quant_dim(b.w);
        __builtin_nontemporal_store(ra, out + i);       // global_store_b128 NT
        __builtin_nontemporal_store(rb, out + i + T);

        i = inext;
        buf ^= 1;
    }
}

// Generic scalar tail (covers any elements not handled by the vector kernel).
__global__ void __launch_bounds__(256)
yijing_tanh_tail(const float* __restrict__ x, float* __restrict__ out,
                 long start, long n) {
    long i = start + (long)blockIdx.x * blockDim.x + threadIdx.x;
    long stride = (long)gridDim.x * blockDim.x;
    for (; i < n; i += stride) {
        out[i] = soft_quant_dim(__builtin_nontemporal_load(x + i));
    }
}

extern "C" void kernel_launch(void* const* d_in, const int* in_sizes, int n_in,
                              void* d_out, int out_size, void* d_ws, size_t ws_size,
                              hipStream_t stream) {
    const float* x = (const float*)d_in[0];
    float* out = (float*)d_out;

    long n  = (long)in_sizes[0];    // 2048*8192*6 = 100,663,296 floats
    long n4 = n >> 2;               // 25,165,824 float4s (hipMalloc base is aligned)

    const int  block = 256;         // 8 wave32s per block
    const int  grid  = 3072;
    const long T     = (long)block * grid;   // 786,432 threads
    const long step  = 2 * T;                // float4s per pipeline stage

    int  iters = (int)(n4 / step);           // 16 exact iterations for this shape
    long main4 = (long)iters * step;

    if (iters > 0) {
        yijing_tanh_stream<<<grid, block, 0, stream>>>(
            (const v4f*)x, (v4f*)out, iters, T);
    }

    long tail_start = main4 << 2;
    if (tail_start < n) {
        long tail = n - tail_start;
        int tblocks = (int)((tail + block - 1) / block);
        if (tblocks > 2048) tblocks = 2048;
        if (tblocks < 1) tblocks = 1;
        yijing_tanh_tail<<<tblocks, block, 0, stream>>>(x, out, tail_start, n);
    }
}